// DGNN_91242285236233
// MI455X (gfx1250) — compile-verified
//
#include <hip/hip_runtime.h>

typedef float v2f __attribute__((ext_vector_type(2)));
typedef float v4f __attribute__((ext_vector_type(4)));
typedef float v8f __attribute__((ext_vector_type(8)));

#define IN_DIM 128
#define HIDDEN 64

// ---------------- degree / norm ----------------
__global__ void k_init_deg(float* __restrict__ deg, int n) {
    int i = blockIdx.x * blockDim.x + threadIdx.x;
    if (i < n) deg[i] = 1.0f;  // self-loop contributes 1
}

__global__ void k_count_deg(const int* __restrict__ rows, float* __restrict__ deg, int e) {
    int i = blockIdx.x * blockDim.x + threadIdx.x;
    if (i < e) atomicAdd(&deg[rows[i]], 1.0f);
}

__global__ void k_rsqrt(float* __restrict__ deg, int n) {
    int i = blockIdx.x * blockDim.x + threadIdx.x;
    if (i < n) deg[i] = rsqrtf(deg[i]);  // deg >= 1 always (self loop)
}

// ---------------- layer-1 GEMM: h = x @ W1  (f32 WMMA 16x16x4) ----------------
// A (16x4 f32): element (m,k) -> lane = m + 16*(k>>1), vgpr = k&1
// B (4x16 f32): element (k,n) -> lane = n + 16*(k>>1), vgpr = k&1
// C/D (16x16 f32): vgpr r, lanes 0-15: (M=r, N=lane); lanes 16-31: (M=r+8, N=lane-16)
__global__ void k_gemm1_wmma(const float* __restrict__ x,
                             const float* __restrict__ w1,
                             float* __restrict__ h,
                             int n, int ntiles) {
    const int lane = threadIdx.x & 31;
    const int wave = threadIdx.x >> 5;
    const int tile = blockIdx.x * (blockDim.x >> 5) + wave;
    if (tile >= ntiles) return;          // wave-uniform: EXEC stays all-1s

    const int m    = lane & 15;
    const int half = lane >> 4;
    const long row0 = (long)tile * 16;

    long arow = row0 + m;
    if (arow >= n) arow = n - 1;         // clamp loads for ragged last tile
    const float* xrow = x + arow * IN_DIM;

    v8f acc0 = {}, acc1 = {}, acc2 = {}, acc3 = {};

    for (int kk = 0; kk < IN_DIM; kk += 4) {
        v2f a = *(const v2f*)(xrow + kk + 2 * half);
        const float* wk0 = w1 + (long)(kk + 2 * half) * HIDDEN;
        const float* wk1 = wk0 + HIDDEN;
        v2f b0 = { wk0[m +  0], wk1[m +  0] };
        v2f b1 = { wk0[m + 16], wk1[m + 16] };
        v2f b2 = { wk0[m + 32], wk1[m + 32] };
        v2f b3 = { wk0[m + 48], wk1[m + 48] };
        acc0 = __builtin_amdgcn_wmma_f32_16x16x4_f32(false, a, false, b0, (short)0, acc0, false, false);
        acc1 = __builtin_amdgcn_wmma_f32_16x16x4_f32(false, a, false, b1, (short)0, acc1, false, false);
        acc2 = __builtin_amdgcn_wmma_f32_16x16x4_f32(false, a, false, b2, (short)0, acc2, false, false);
        acc3 = __builtin_amdgcn_wmma_f32_16x16x4_f32(false, a, false, b3, (short)0, acc3, false, false);
    }

    #pragma unroll
    for (int r = 0; r < 8; ++r) {
        long mrow = row0 + r + 8 * half;
        if (mrow < n) {
            float* hr = h + mrow * HIDDEN;
            hr[m +  0] = acc0[r];
            hr[m + 16] = acc1[r];
            hr[m + 32] = acc2[r];
            hr[m + 48] = acc3[r];
        }
    }
}

// ---------------- self-loop init of agg (also clears poison) ----------------
__global__ void k_self_msg(const float* __restrict__ h, const float* __restrict__ dinv,
                           float* __restrict__ agg, int n) {
    int t = blockIdx.x * blockDim.x + threadIdx.x;   // one per (node, 4 dims)
    int total = n * (HIDDEN / 4);
    if (t >= total) return;
    int node = t >> 4;
    float s = dinv[node]; s = s * s;
    v4f v = ((const v4f*)h)[t];
    v4f o = { v.x * s, v.y * s, v.z * s, v.w * s };
    ((v4f*)agg)[t] = o;
}

// ---------------- edge scatter, layer 1: wave per edge, 2 dims/lane ----------------
__global__ void k_scatter1(const int* __restrict__ rows, const int* __restrict__ cols,
                           const float* __restrict__ h, const float* __restrict__ dinv,
                           float* __restrict__ agg, int e) {
    int gw   = (int)(((long)blockIdx.x * blockDim.x + threadIdx.x) >> 5);
    int lane = threadIdx.x & 31;
    if (gw >= e) return;
    int d = rows[gw], s = cols[gw];
    float norm = dinv[d] * dinv[s];
    int dim = lane * 2;
    v2f hv = *(const v2f*)(h + (long)s * HIDDEN + dim);
    atomicAdd(agg + (long)d * HIDDEN + dim,     hv.x * norm);
    atomicAdd(agg + (long)d * HIDDEN + dim + 1, hv.y * norm);
}

// ---------------- relu + bias + layer-2 transform + out init ----------------
__global__ void k_relu_dot(const float* __restrict__ agg, const float* __restrict__ b1,
                           const float* __restrict__ w2, const float* __restrict__ b2,
                           const float* __restrict__ dinv,
                           float* __restrict__ z, float* __restrict__ out, int n) {
    int gw   = (int)(((long)blockIdx.x * blockDim.x + threadIdx.x) >> 5);
    int lane = threadIdx.x & 31;
    if (gw >= n) return;
    int dim = lane * 2;
    float v0 = agg[(long)gw * HIDDEN + dim]     + b1[dim];
    float v1 = agg[(long)gw * HIDDEN + dim + 1] + b1[dim + 1];
    v0 = fmaxf(v0, 0.0f);
    v1 = fmaxf(v1, 0.0f);
    float sum = v0 * w2[dim] + v1 * w2[dim + 1];
    #pragma unroll
    for (int off = 16; off > 0; off >>= 1)
        sum += __shfl_xor(sum, off, 32);
    if (lane == 0) {
        z[gw] = sum;
        float di = dinv[gw];
        out[gw] = b2[0] + di * di * sum;     // self-loop term + bias
    }
}

// ---------------- edge scatter, layer 2: thread per edge ----------------
__global__ void k_scatter2(const int* __restrict__ rows, const int* __restrict__ cols,
                           const float* __restrict__ z, const float* __restrict__ dinv,
                           float* __restrict__ out, int e) {
    int i = blockIdx.x * blockDim.x + threadIdx.x;
    if (i < e) {
        int d = rows[i], s = cols[i];
        atomicAdd(&out[d], z[s] * dinv[d] * dinv[s]);
    }
}

extern "C" void kernel_launch(void* const* d_in, const int* in_sizes, int n_in,
                              void* d_out, int out_size, void* d_ws, size_t ws_size,
                              hipStream_t stream) {
    const float* x  = (const float*)d_in[0];
    const float* W1 = (const float*)d_in[1];
    const float* b1 = (const float*)d_in[2];
    const float* W2 = (const float*)d_in[3];
    const float* b2 = (const float*)d_in[4];
    const int*   ei = (const int*)d_in[5];

    const int N = in_sizes[0] / IN_DIM;
    const int E = in_sizes[5] / 2;
    const int* rows = ei;        // edge_index[0] = destinations
    const int* cols = ei + E;    // edge_index[1] = sources

    char* ws = (char*)d_ws;
    size_t szN   = (((size_t)N * 4)            + 255) & ~(size_t)255;
    size_t szNH  = (((size_t)N * HIDDEN * 4)   + 255) & ~(size_t)255;
    float* deg = (float*)(ws);                 // becomes dinv in-place
    float* z   = (float*)(ws + szN);
    float* h   = (float*)(ws + 2 * szN);
    float* agg = (float*)(ws + 2 * szN + szNH);

    const int T = 256;

    k_init_deg <<<(N + T - 1) / T, T, 0, stream>>>(deg, N);
    k_count_deg<<<(E + T - 1) / T, T, 0, stream>>>(rows, deg, E);
    k_rsqrt    <<<(N + T - 1) / T, T, 0, stream>>>(deg, N);

    const int ntiles = (N + 15) / 16;
    const int wpb = 4;  // 4 waves / 128 threads per block
    k_gemm1_wmma<<<(ntiles + wpb - 1) / wpb, wpb * 32, 0, stream>>>(x, W1, h, N, ntiles);

    int tot5 = N * (HIDDEN / 4);
    k_self_msg<<<(tot5 + T - 1) / T, T, 0, stream>>>(h, deg, agg, N);

    long th6 = (long)E * 32;
    k_scatter1<<<(int)((th6 + T - 1) / T), T, 0, stream>>>(rows, cols, h, deg, agg, E);

    long th7 = (long)N * 32;
    k_relu_dot<<<(int)((th7 + T - 1) / T), T, 0, stream>>>(agg, b1, W2, b2, deg, z, (float*)d_out, N);

    k_scatter2<<<(E + T - 1) / T, T, 0, stream>>>(rows, cols, z, deg, (float*)d_out, E);
}